// TransientLoss_3384434229475
// MI455X (gfx1250) — compile-verified
//
#include <hip/hip_runtime.h>

// ---------------- problem constants (from reference) ----------------
#define SR_B   16
#define SR_N   480000
#define NOUT   (SR_N + 1)                    // 480001 envelope samples per row
#define WINL   240
#define PADL   120
#define EPSF   1e-6f

// ---------------- tiling ----------------
#define NTHR   256                           // 8 waves (wave32)
#define GPT    16                            // envelope outputs per thread
#define TILE   (NTHR * GPT)                  // 4096 outputs per block
#define DLEN   (TILE + WINL)                 // 4336 d-values needed per tile
#define TPR    ((NOUT + TILE - 1) / TILE)    // 118 tiles per row
#define NBLK   (SR_B * TPR)                  // 1888 blocks

// LDS bank-conflict-free layout for the d buffers: pad 1 float per 16.
// Per-lane stride becomes 17 floats; gcd(17,64)=1 -> conflict-free.
#define PIDX(l)  ((l) + ((l) >> 4))
#define DPLEN    (DLEN + (DLEN >> 4) + 1)    // 4336 + 271 + 1
#define XLEN     4344                        // DLEN+1+3 rounded up to mult of 4

#if __has_builtin(__builtin_amdgcn_global_load_async_to_lds_b32) && \
    __has_builtin(__builtin_amdgcn_s_wait_asynccnt)
#define HAVE_ASYNC_LDS 1
typedef __attribute__((address_space(1))) int* gaddr_t;
typedef __attribute__((address_space(3))) int* laddr_t;
#if __has_builtin(__builtin_amdgcn_global_load_async_to_lds_b128)
#define HAVE_ASYNC_B128 1
typedef int v4i __attribute__((ext_vector_type(4)));
typedef __attribute__((address_space(1))) v4i* gaddr4_t;
typedef __attribute__((address_space(3))) v4i* laddr4_t;
#endif
#endif

// Stage the d-array (|x[i]-x[i-1]|, zero-padded outside [1,N)) for one tile
// into LDS (padded layout). dbuf[PIDX(l)] = d[j0 - 120 + l], l in [0, DLEN).
// Ends with a barrier: dbuf fully valid, xbuf free for reuse.
__device__ __forceinline__ void fill_d(const float* __restrict__ xrow, int j0,
                                       float* __restrict__ dbuf,
                                       float* __restrict__ xbuf) {
  const int t  = threadIdx.x;
  const int lo = j0 - PADL - 1;              // x index mapped to xbuf[delta]
#if defined(HAVE_ASYNC_LDS)
  const int lo4   = lo & ~3;                 // 16B-aligned global start
  const int delta = lo - lo4;                // 0..3
  const int nc    = (DLEN + 1 + delta + 3) >> 2;  // #b128 chunks
  // interior tile: whole aligned range [lo4, lo4 + 4*nc) inside the row
  const bool interior = (lo4 >= 0) && (lo4 + 4 * nc <= SR_N);
  if (interior) {
    // CDNA5 async global -> LDS copy (ASYNCcnt-tracked, no VGPR round trip)
#if defined(HAVE_ASYNC_B128)
    for (int c = t; c < nc; c += NTHR) {
      __builtin_amdgcn_global_load_async_to_lds_b128(
          (gaddr4_t)(xrow + lo4 + 4 * c),
          (laddr4_t)(xbuf + 4 * c), 0, 0);
    }
#else
    for (int l = t; l < 4 * nc; l += NTHR) {
      __builtin_amdgcn_global_load_async_to_lds_b32(
          (gaddr_t)(xrow + lo4 + l),
          (laddr_t)(xbuf + l), 0, 0);
    }
#endif
    __builtin_amdgcn_s_wait_asynccnt(0);     // my wave's async copies done
    __syncthreads();                         // everyone's copies visible
    for (int l = t; l < DLEN; l += NTHR)
      dbuf[PIDX(l)] = fabsf(xbuf[l + delta + 1] - xbuf[l + delta]);
    __syncthreads();
    return;
  }
#endif
  // edge tiles (or no async builtin): guarded scalar path
  for (int l = t; l < DLEN; l += NTHR) {
    const int i = j0 - PADL + l;             // global d index
    float dv = 0.0f;
    if (i >= 1 && i < SR_N) dv = fabsf(xrow[i] - xrow[i - 1]);
    dbuf[PIDX(l)] = dv;
  }
  __syncthreads();
}

// ---------------- kernel: weighted log-envelope loss, per-block partial ------
__global__ __launch_bounds__(NTHR)
void k_loss(const float* __restrict__ yp, const float* __restrict__ yt,
            const unsigned int* __restrict__ rowmax,
            float* __restrict__ partial) {
  __shared__ __align__(16) float xbuf[XLEN];
  __shared__ float dp[DPLEN];
  __shared__ float dt[DPLEN];
  __shared__ float red[NTHR];
  const int blk = blockIdx.x;
  const int row = blk / TPR;
  const int j0  = (blk % TPR) * TILE;

  fill_d(yp + (size_t)row * SR_N, j0, dp, xbuf);
  fill_d(yt + (size_t)row * SR_N, j0, dt, xbuf);

  const float invr = 1.0f / (__uint_as_float(rowmax[row]) + EPSF);
  const int l0 = threadIdx.x * GPT;

  float sp = 0.0f, st = 0.0f;
  #pragma unroll 8
  for (int m = 0; m < WINL; ++m) {
    sp += dp[PIDX(l0 + m)];
    st += dt[PIDX(l0 + m)];
  }

  float acc = 0.0f;
  #pragma unroll
  for (int g = 0; g < GPT; ++g) {
    const int j = j0 + l0 + g;
    if (j < NOUT) {
      const float ep = sp * (1.0f / WINL);
      const float et = st * (1.0f / WINL);
      const float dl = fabsf(__logf(ep + EPSF) - __logf(et + EPSF));
      acc += dl * (0.2f + 0.8f * (et * invr));
    }
    sp += dp[PIDX(l0 + g + WINL)] - dp[PIDX(l0 + g)];   // slide the window
    st += dt[PIDX(l0 + g + WINL)] - dt[PIDX(l0 + g)];
  }

  red[threadIdx.x] = acc;
  __syncthreads();
  for (int off = NTHR >> 1; off > 0; off >>= 1) {
    if (threadIdx.x < (unsigned)off)
      red[threadIdx.x] += red[threadIdx.x + off];
    __syncthreads();
  }
  if (threadIdx.x == 0) partial[blk] = red[0];   // per-block slot: deterministic
}

// ---------------- kernel: per-row max of the true envelope ----------------
__global__ __launch_bounds__(NTHR)
void k_rowmax(const float* __restrict__ yt, unsigned int* __restrict__ rowmax) {
  __shared__ __align__(16) float xbuf[XLEN];
  __shared__ float dt[DPLEN];
  __shared__ float red[NTHR];
  const int blk = blockIdx.x;
  const int row = blk / TPR;
  const int j0  = (blk % TPR) * TILE;

  fill_d(yt + (size_t)row * SR_N, j0, dt, xbuf);

  const int l0 = threadIdx.x * GPT;
  float s = 0.0f;
  #pragma unroll 8
  for (int m = 0; m < WINL; ++m) s += dt[PIDX(l0 + m)];

  float mx = 0.0f;
  #pragma unroll
  for (int g = 0; g < GPT; ++g) {
    const int j = j0 + l0 + g;
    if (j < NOUT) mx = fmaxf(mx, s * (1.0f / WINL));
    s += dt[PIDX(l0 + g + WINL)] - dt[PIDX(l0 + g)];
  }

  red[threadIdx.x] = mx;
  __syncthreads();
  for (int off = NTHR >> 1; off > 0; off >>= 1) {
    if (threadIdx.x < (unsigned)off)
      red[threadIdx.x] = fmaxf(red[threadIdx.x], red[threadIdx.x + off]);
    __syncthreads();
  }
  // envelopes are >= 0, so float-as-uint atomicMax is order-preserving
  if (threadIdx.x == 0) atomicMax(&rowmax[row], __float_as_uint(red[0]));
}

// ---------------- kernel: final fixed-order reduction ----------------
__global__ __launch_bounds__(NTHR)
void k_final(const float* __restrict__ partial, float* __restrict__ out) {
  __shared__ float red[NTHR];
  float s = 0.0f;
  for (int i = threadIdx.x; i < NBLK; i += NTHR) s += partial[i];
  red[threadIdx.x] = s;
  __syncthreads();
  for (int off = NTHR >> 1; off > 0; off >>= 1) {
    if (threadIdx.x < (unsigned)off)
      red[threadIdx.x] += red[threadIdx.x + off];
    __syncthreads();
  }
  if (threadIdx.x == 0)
    out[0] = red[0] / (float)((long long)SR_B * (long long)NOUT);
}

// ---------------- kernel: zero the per-row maxima ----------------
__global__ void k_init(unsigned int* __restrict__ rowmax) {
  if (threadIdx.x < SR_B) rowmax[threadIdx.x] = 0u;
}

// ---------------- host entry ----------------
extern "C" void kernel_launch(void* const* d_in, const int* in_sizes, int n_in,
                              void* d_out, int out_size, void* d_ws, size_t ws_size,
                              hipStream_t stream) {
  const float* yp = (const float*)d_in[0];   // y_pred (16, 480000) f32
  const float* yt = (const float*)d_in[1];   // y_true (16, 480000) f32
  float* out = (float*)d_out;

  unsigned int* rowmax  = (unsigned int*)d_ws;           // 16 u32
  float*        partial = (float*)d_ws + 16;             // NBLK f32

  k_init  <<<1,    64,   0, stream>>>(rowmax);
  k_rowmax<<<NBLK, NTHR, 0, stream>>>(yt, rowmax);
  k_loss  <<<NBLK, NTHR, 0, stream>>>(yp, yt, rowmax, partial);
  k_final <<<1,    NTHR, 0, stream>>>(partial, out);
}